// GFR_34256659153565
// MI455X (gfx1250) — compile-verified
//
#include <hip/hip_runtime.h>
#include <math.h>

// ---------------------------------------------------------------------------
// GFR neuron recurrence on MI455X (gfx1250, wave32).
//
// One wave = 16 traces. The 16 filters x 16 traces state matrix V lives in the
// WMMA f32 B-matrix layout: 4 chunks (K=4) of 2 VGPRs; slot (chunk c, comp j,
// laneHalf h) <-> filter f = c*4 + j*2 + h (our bijection; the WMMA column sum
// is invariant to it). Column (trace) = lane % 16 for both B operand and C/D
// result, so each lane's D[0] is the mean of its own trace.
//
// mean_over_filters = (1/16 * ones(16x4)) x V_chunk accumulated over 4 chunks
// with V_WMMA_F32_16X16X4_F32. tanh via v_tanh_f32 (TRANS).
//
// Stores: timesteps processed in pairs; lane-half 0 writes step t, lane-half 1
// writes step t+1 (same trace, identical redundant fs) -> uniform EXEC, no
// saveexec/branch on the serial chain, half the store count.
// ---------------------------------------------------------------------------

typedef float v2f __attribute__((ext_vector_type(2)));
typedef float v8f __attribute__((ext_vector_type(8)));

#define T_STEPS 1024
#define N_FILT 16
#define TRACES_PER_WAVE 16
#define WAVES_PER_BLOCK 8
#define TRACES_PER_BLOCK (TRACES_PER_WAVE * WAVES_PER_BLOCK)   // 128
#define THREADS_PER_BLOCK 256

__global__ __launch_bounds__(THREADS_PER_BLOCK, 2)
void gfr_recurrence_kernel(const float* __restrict__ currents,   // (B, T)
                           const float* __restrict__ a,          // (1, 16)
                           const float* __restrict__ b,          // (1, 16)
                           const float* __restrict__ poly_coeff, // (1, 5)
                           const float* __restrict__ g_b,        // (1,)
                           const float* __restrict__ ds,         // (16,)
                           const float* __restrict__ max_current,     // (1,)
                           const float* __restrict__ max_firing_rate, // (1,)
                           float* __restrict__ out,              // (T, B)
                           int Btot)
{
    const int lane = threadIdx.x & 31;
    const int wave = threadIdx.x >> 5;
    const int col  = lane & 15;   // trace column within the wave's 16-trace tile
    const int half = lane >> 4;   // filter-parity half; also the store phase

    int trace = blockIdx.x * TRACES_PER_BLOCK + wave * TRACES_PER_WAVE + col;
    const bool valid = (trace < Btot);
    if (trace >= Btot) trace = Btot - 1;   // clamp for memory safety

    // Per-slot filter constants under our slot->filter bijection.
    float dec[4][2], am[4][2], bm[4][2];
#pragma unroll
    for (int c = 0; c < 4; ++c) {
#pragma unroll
        for (int j = 0; j < 2; ++j) {
            const int f = c * 4 + j * 2 + half;
            dec[c][j] = 1.0f - ds[f];
            am[c][j]  = a[f];
            bm[c][j]  = 1000.0f * b[f];   // fold the 1000 from the reference
        }
    }

    // Squared polynomial coefficients (Horner order).
    const float c0q = poly_coeff[0] * poly_coeff[0];
    const float c1q = poly_coeff[1] * poly_coeff[1];
    const float c2q = poly_coeff[2] * poly_coeff[2];
    const float c3q = poly_coeff[3] * poly_coeff[3];
    const float c4q = poly_coeff[4] * poly_coeff[4];

    const float inv   = 1.0f / max_current[0];
    const float xbias = -g_b[0] * inv;          // x = mean*inv + xbias
    const float mfr   = max_firing_rate[0];

    // State: V in WMMA B-matrix layout, 4 chunks of v2f. fs feedback scalar.
    v2f vS[4];
#pragma unroll
    for (int c = 0; c < 4; ++c) { vS[c].x = 0.0f; vS[c].y = 0.0f; }
    float fs = 0.0f;

    // Constant A matrix = 1/16 everywhere -> D = mean over all 16 filters.
    v2f aOnes; aOnes.x = 0.0625f; aOnes.y = 0.0625f;

    // Per-lane contiguous current row; float4 software double buffer.
    const float4* __restrict__ row4 =
        (const float4*)(currents + (size_t)trace * T_STEPS);
    float4 nxt = row4[0];

    for (int tb = 0; tb < T_STEPS; tb += 4) {
        const float4 cur4 = nxt;
        int ni = (tb >> 2) + 1;
        if (ni > (T_STEPS / 4) - 1) ni = (T_STEPS / 4) - 1;
        nxt = row4[ni];   // prefetch next 4 timesteps off the critical path

        const float curArr[4] = {cur4.x, cur4.y, cur4.z, cur4.w};

#pragma unroll
        for (int k = 0; k < 2; ++k) {            // two timestep-pairs per chunk
            float fsPair[2];
#pragma unroll
            for (int i = 0; i < 2; ++i) {
                const float cur = curArr[2 * k + i];

                // v = decay*v + (cur*a + (1000*fs)*b), element-wise per slot.
#pragma unroll
                for (int c = 0; c < 4; ++c) {
                    vS[c].x = fmaf(dec[c][0], vS[c].x, fmaf(fs, bm[c][0], cur * am[c][0]));
                    vS[c].y = fmaf(dec[c][1], vS[c].y, fmaf(fs, bm[c][1], cur * am[c][1]));
                }

                // Mean over filters via chained 16x16x4 f32 WMMA column-sums.
                float mean;
#if __has_builtin(__builtin_amdgcn_wmma_f32_16x16x4_f32)
                v8f acc = {};
                acc = __builtin_amdgcn_wmma_f32_16x16x4_f32(
                    false, aOnes, false, vS[0], (short)0, acc, false, false);
                acc = __builtin_amdgcn_wmma_f32_16x16x4_f32(
                    false, aOnes, false, vS[1], (short)0, acc, false, false);
                acc = __builtin_amdgcn_wmma_f32_16x16x4_f32(
                    false, aOnes, false, vS[2], (short)0, acc, false, false);
                acc = __builtin_amdgcn_wmma_f32_16x16x4_f32(
                    false, aOnes, false, vS[3], (short)0, acc, false, false);
                mean = acc[0];   // replicated across rows; lane's trace column
#else
                float s = (vS[0].x + vS[0].y) + (vS[1].x + vS[1].y) +
                          (vS[2].x + vS[2].y) + (vS[3].x + vS[3].y);
                s += __shfl_xor(s, 16, 32);
                mean = s * 0.0625f;
#endif

                // Polynomial activation: relu(mfr * tanh(poly(x))).
                const float x = fmaf(mean, inv, xbias);
                const float p = fmaf(fmaf(fmaf(fmaf(c4q, x, c3q), x, c2q), x, c1q), x, c0q);
                float th;
#if __has_builtin(__builtin_amdgcn_tanhf)
                th = __builtin_amdgcn_tanhf(p);   // v_tanh_f32, 1 TRANS op
#else
                th = tanhf(p);
#endif
                fs = fmaxf(0.0f, mfr * th);
                fsPair[i] = fs;
            }

            // Uniform store: lane-half 0 -> step t, lane-half 1 -> step t+1.
            // Every (t, trace) written exactly once; EXEC stays full.
            if (valid) {
                const float fsSel = half ? fsPair[1] : fsPair[0];
                out[(size_t)(tb + 2 * k + half) * (size_t)Btot + (size_t)trace] = fsSel;
            }
        }
    }
}

extern "C" void kernel_launch(void* const* d_in, const int* in_sizes, int n_in,
                              void* d_out, int out_size, void* d_ws, size_t ws_size,
                              hipStream_t stream) {
    const float* currents        = (const float*)d_in[0];
    const float* a               = (const float*)d_in[1];
    const float* b               = (const float*)d_in[2];
    const float* poly_coeff      = (const float*)d_in[3];
    const float* g_b             = (const float*)d_in[4];
    const float* ds              = (const float*)d_in[5];
    const float* max_current     = (const float*)d_in[6];
    const float* max_firing_rate = (const float*)d_in[7];
    float* out = (float*)d_out;

    const int Btot = in_sizes[0] / T_STEPS;                             // 4096
    const int grid = (Btot + TRACES_PER_BLOCK - 1) / TRACES_PER_BLOCK;  // 32

    gfr_recurrence_kernel<<<grid, THREADS_PER_BLOCK, 0, stream>>>(
        currents, a, b, poly_coeff, g_b, ds, max_current, max_firing_rate,
        out, Btot);
}